// HeteroGraphSAGEModel_39934605918237
// MI455X (gfx1250) — compile-verified
//
#include <hip/hip_runtime.h>

typedef __attribute__((ext_vector_type(2))) float v2f;
typedef __attribute__((ext_vector_type(8))) float v8f;

#define HID 128

__global__ __launch_bounds__(256) void zero_k(float* __restrict__ p, long n) {
  long i = (long)blockIdx.x * blockDim.x + threadIdx.x;
  long stride = (long)gridDim.x * blockDim.x;
  for (; i < n; i += stride) p[i] = 0.f;
}

__global__ __launch_bounds__(256) void relu_k(float* __restrict__ p, long n) {
  long i = (long)blockIdx.x * blockDim.x + threadIdx.x;
  long stride = (long)gridDim.x * blockDim.x;
  for (; i < n; i += stride) p[i] = fmaxf(p[i], 0.f);
}

__global__ __launch_bounds__(256) void count_k(const int* __restrict__ dst,
                                               float* __restrict__ cnt, int E) {
  int i = blockIdx.x * blockDim.x + threadIdx.x;
  if (i < E) atomicAdd(cnt + dst[i], 1.0f);
}

// Sum Wl[t] / b[t] over the relations whose dst is this node type.
// blockIdx.x = node type (0 user, 1 product, 2 category, 3 query)
__global__ __launch_bounds__(256) void wsum_k(const float* __restrict__ Wl,
                                              const float* __restrict__ b,
                                              float* __restrict__ Wsum,
                                              float* __restrict__ bsum) {
  const int masks[4] = {0x0A, 0x91, 0x40, 0x24};  // user{1,3} product{0,4,7} category{6} query{2,5}
  int mask = masks[blockIdx.x];
  float* wo = Wsum + (long)blockIdx.x * (HID * HID);
  for (int i = threadIdx.x; i < HID * HID; i += blockDim.x) {
    float s = 0.f;
#pragma unroll
    for (int t = 0; t < 8; ++t)
      if ((mask >> t) & 1) s += Wl[t * HID * HID + i];
    wo[i] = s;
  }
  if (threadIdx.x < HID) {
    float s = 0.f;
#pragma unroll
    for (int t = 0; t < 8; ++t)
      if ((mask >> t) & 1) s += b[t * HID + threadIdx.x];
    bsum[blockIdx.x * HID + threadIdx.x] = s;
  }
}

// Y[M,128] = X[M,128] * W^T (+ bias), W row-major [128,128].
// Block = 32 rows; wave w owns cols [16w,16w+16), two 16x16 accumulators.
// Uses V_WMMA_F32_16X16X4_F32 (f32 A/B, f32 accum) — exact-precision path.
__global__ __launch_bounds__(256) void gemm_xwT_k(const float* __restrict__ X,
                                                  const float* __restrict__ W,
                                                  const float* __restrict__ bias,
                                                  float* __restrict__ Y, int M) {
  int lane = threadIdx.x & 31;
  int wv   = threadIdx.x >> 5;
  int m0 = blockIdx.x * 32;
  int m1 = m0 + 16;
  bool has2 = (m1 < M);          // block-uniform; EXEC stays all-ones
  int n0 = wv * 16;
  int li = lane & 15;
  int kh = (lane >> 4) << 1;     // 0 (lanes 0-15) or 2 (lanes 16-31)
  int mrA = m0 + li;
  int mrB = (has2 ? m1 : m0) + li;   // clamped for safe reads
  int nn  = n0 + li;
  const float* xa = X + (long)mrA * HID;
  const float* xb = X + (long)mrB * HID;
  const float* wr = W + (long)nn * HID;   // row nn of W == col nn of W^T
  float bv = bias ? bias[nn] : 0.f;
  v8f c0, c1;
#pragma unroll
  for (int i = 0; i < 8; ++i) { c0[i] = bv; c1[i] = bv; }
#pragma unroll
  for (int k0 = 0; k0 < HID; k0 += 4) {
    v2f bf = *(const v2f*)(wr + k0 + kh);   // B: K rows {k0+kh, k0+kh+1}, N=nn
    v2f a0 = *(const v2f*)(xa + k0 + kh);   // A: M=mrA, K={k0+kh, k0+kh+1}
    v2f a1 = *(const v2f*)(xb + k0 + kh);
    c0 = __builtin_amdgcn_wmma_f32_16x16x4_f32(false, a0, false, bf, (short)0, c0, false, false);
    c1 = __builtin_amdgcn_wmma_f32_16x16x4_f32(false, a1, false, bf, (short)0, c1, false, false);
  }
  int rb0 = m0 + ((lane >> 4) << 3);        // C/D: VGPR v -> row v (+8 for hi lanes)
#pragma unroll
  for (int v = 0; v < 8; ++v) Y[(long)(rb0 + v) * HID + nn] = c0[v];
  if (has2) {
    int rb1 = m1 + ((lane >> 4) << 3);
#pragma unroll
    for (int v = 0; v < 8; ++v) Y[(long)(rb1 + v) * HID + nn] = c1[v];
  }
}

// Pre-divided scatter-mean: out[dst] += T[src] * w / max(cnt[dst],1)
// One wave per edge (512B coalesced gather), 4 channels/lane.
__global__ __launch_bounds__(256) void scatter_k(const float* __restrict__ T,
                                                 const int* __restrict__ src,
                                                 const int* __restrict__ dst,
                                                 const float* __restrict__ w,
                                                 const float* __restrict__ cnt,
                                                 float* __restrict__ out, int E) {
  int e = blockIdx.x * 8 + (threadIdx.x >> 5);
  if (e >= E) return;
  int lane = threadIdx.x & 31;
  int s = src[e];
  int d = dst[e];
  float scale = w[e] / fmaxf(cnt[d], 1.0f);
  float4 v = *((const float4*)(T + (long)s * HID) + lane);
  float* o = out + (long)d * HID + lane * 4;
  atomicAdd(o + 0, v.x * scale);
  atomicAdd(o + 1, v.y * scale);
  atomicAdd(o + 2, v.z * scale);
  atomicAdd(o + 3, v.w * scale);
}

extern "C" void kernel_launch(void* const* d_in, const int* in_sizes, int n_in,
                              void* d_out, int out_size, void* d_ws, size_t ws_size,
                              hipStream_t stream) {
  const float* emb[4] = {(const float*)d_in[0], (const float*)d_in[1],
                         (const float*)d_in[2], (const float*)d_in[3]};
  const float* W1l = (const float*)d_in[4];
  const float* b1  = (const float*)d_in[5];
  const float* W1r = (const float*)d_in[6];
  const float* W2l = (const float*)d_in[7];
  const float* b2  = (const float*)d_in[8];
  const float* W2r = (const float*)d_in[9];
  const float* ew[8];
  for (int i = 0; i < 8; ++i) ew[i] = (const float*)d_in[10 + i];
  const int* ei[4] = {(const int*)d_in[18], (const int*)d_in[19],
                      (const int*)d_in[20], (const int*)d_in[21]};
  const int EN[4] = {500000, 300000, 300000, 100000};

  // node types: 0 user, 1 product, 2 category, 3 query (== output order)
  const int  rowsT[4]  = {200000, 100000, 2000, 50000};
  const long rowOff[4] = {0, 200000, 300000, 302000};
  const long NTOT = 352000;

  struct RelH { int st, dt; const int* s; const int* d; const float* w; int E; };
  RelH rel[8];
  const int fwdPair[4][2] = {{0, 1}, {0, 3}, {3, 1}, {1, 2}};  // buys, searches, matches, in
  for (int i = 0; i < 4; ++i) {
    const int* e = ei[i];
    int E = EN[i];
    rel[2 * i + 0] = RelH{fwdPair[i][0], fwdPair[i][1], e,     e + E, ew[2 * i + 0], E};
    rel[2 * i + 1] = RelH{fwdPair[i][1], fwdPair[i][0], e + E, e,     ew[2 * i + 1], E};
  }

  float* ws = (float*)d_ws;
  float* out = (float*)d_out;
  const long H_OFF   = 0;                       // hidden layer, 45,056,000 f
  const long TMP_OFF = NTOT * HID;              // transformed src, 25,600,000 f
  const long CNT_OFF = TMP_OFF + 200000L * HID;
  long cntOff[8];
  long c = CNT_OFF;
  for (int t = 0; t < 8; ++t) { cntOff[t] = c; c += rowsT[rel[t].dt]; }
  const long WS_OFF = c;                        // 2 layers x 4 types x 128x128
  const long BS_OFF = WS_OFF + 2L * 4 * HID * HID;
  float* h   = ws + H_OFF;
  float* tmp = ws + TMP_OFF;

  // degree counts (shared by both layers)
  zero_k<<<1024, 256, 0, stream>>>(ws + CNT_OFF, c - CNT_OFF);
  for (int t = 0; t < 8; ++t)
    count_k<<<(rel[t].E + 255) / 256, 256, 0, stream>>>(rel[t].d, ws + cntOff[t], rel[t].E);

  // per-dst-type summed Wl / b for both layers
  wsum_k<<<4, 256, 0, stream>>>(W1l, b1, ws + WS_OFF, ws + BS_OFF);
  wsum_k<<<4, 256, 0, stream>>>(W2l, b2, ws + WS_OFF + 4L * HID * HID, ws + BS_OFF + 4L * HID);

  // ---- layer 1: h = sum_t [ scatter-mean + x@Wl^T + b ] ----
  for (int d = 0; d < 4; ++d)
    gemm_xwT_k<<<(rowsT[d] + 31) / 32, 256, 0, stream>>>(
        emb[d], ws + WS_OFF + (long)d * HID * HID, ws + BS_OFF + (long)d * HID,
        h + rowOff[d] * HID, rowsT[d]);
  for (int t = 0; t < 8; ++t) {
    gemm_xwT_k<<<(rowsT[rel[t].st] + 31) / 32, 256, 0, stream>>>(
        emb[rel[t].st], W1r + (long)t * HID * HID, nullptr, tmp, rowsT[rel[t].st]);
    scatter_k<<<(rel[t].E + 7) / 8, 256, 0, stream>>>(
        tmp, rel[t].s, rel[t].d, rel[t].w, ws + cntOff[t],
        h + rowOff[rel[t].dt] * HID, rel[t].E);
  }
  relu_k<<<2048, 256, 0, stream>>>(h, NTOT * HID);

  // ---- layer 2: out = sum_t [ scatter-mean + h@Wl^T + b ] ----
  for (int d = 0; d < 4; ++d)
    gemm_xwT_k<<<(rowsT[d] + 31) / 32, 256, 0, stream>>>(
        h + rowOff[d] * HID, ws + WS_OFF + (4L + d) * HID * HID, ws + BS_OFF + (4L + d) * HID,
        out + rowOff[d] * HID, rowsT[d]);
  for (int t = 0; t < 8; ++t) {
    gemm_xwT_k<<<(rowsT[rel[t].st] + 31) / 32, 256, 0, stream>>>(
        h + rowOff[rel[t].st] * HID, W2r + (long)t * HID * HID, nullptr, tmp, rowsT[rel[t].st]);
    scatter_k<<<(rel[t].E + 7) / 8, 256, 0, stream>>>(
        tmp, rel[t].s, rel[t].d, rel[t].w, ws + cntOff[t],
        out + rowOff[rel[t].dt] * HID, rel[t].E);
  }
}